// Policy_31894427140496
// MI455X (gfx1250) — compile-verified
//
#include <hip/hip_runtime.h>
#include <hip/hip_bf16.h>

// ---------------- problem constants ----------------
#define TT   16
#define NNB  1024
#define RR   (TT*NNB)      // 16384 rows
#define DD   128
#define OBSZ 578
#define CHK  2048          // rows per K/V chunk

typedef float v2f __attribute__((ext_vector_type(2)));
typedef float v8f __attribute__((ext_vector_type(8)));

#define WMMA_F32(a, b, c) \
    __builtin_amdgcn_wmma_f32_16x16x4_f32(false, (a), false, (b), (short)0, (c), false, false)

// =====================================================================
// Generic fp32 WMMA GEMM:  C[M x N] = A(M x K) * B(K x N)  (+bias)(+relu)
// B is pre-packed K x N row-major (i.e. W transposed, zero-padded rows).
// One wave computes a 32(M) x 64(N) strip: each B fragment is reused by
// two v_wmma_f32_16x16x4_f32 ops (A-fragment pair), ~10.7 FLOP/byte.
// amode==0: A row m at A + m*lda
// amode==1: A row m is entity data: inputs + (cbase+m/S)*rowstride + tbase + (m%S)*ksz
//           (weight rows >= ksz are zero, so over-read lanes contribute 0)
// =====================================================================
__global__ __launch_bounds__(32)
void gemm32x64_f32(const float* __restrict__ A, int lda, int amode,
                   int S, int rowstride, int tbase, int ksz, int cbase,
                   const float* __restrict__ B, int Nn,
                   float* __restrict__ C, int ldc, int coff,
                   const float* __restrict__ bias, int act, int K)
{
    const int l     = threadIdx.x;
    const int lo    = l & 15;
    const int khalf = l >> 4;
    const int m0    = blockIdx.x * 32 + lo;
    const int m1    = m0 + 16;

    const float *arow0, *arow1;
    if (amode == 0) {
        arow0 = A + (size_t)m0 * lda;
        arow1 = A + (size_t)m1 * lda;
    } else {
        arow0 = A + (size_t)(cbase + m0 / S) * rowstride + tbase + (m0 % S) * ksz;
        arow1 = A + (size_t)(cbase + m1 / S) * rowstride + tbase + (m1 % S) * ksz;
    }
    const int ncol = blockIdx.y * 64 + lo;

    v8f p0 = {}, p1 = {}, p2 = {}, p3 = {};   // rows m0-tile
    v8f q0 = {}, q1 = {}, q2 = {}, q3 = {};   // rows m1-tile
    for (int k0 = 0; k0 < K; k0 += 4) {
        const int ka = k0 + 2 * khalf;
        v2f a0; a0.x = arow0[ka]; a0.y = arow0[ka + 1];
        v2f a1; a1.x = arow1[ka]; a1.y = arow1[ka + 1];
        const float* b0p = B + (size_t)ka * Nn + ncol;
        const float* b1p = b0p + Nn;
        v2f b0; b0.x = b0p[0];  b0.y = b1p[0];
        v2f b1; b1.x = b0p[16]; b1.y = b1p[16];
        v2f b2; b2.x = b0p[32]; b2.y = b1p[32];
        v2f b3; b3.x = b0p[48]; b3.y = b1p[48];
        p0 = WMMA_F32(a0, b0, p0);  q0 = WMMA_F32(a1, b0, q0);
        p1 = WMMA_F32(a0, b1, p1);  q1 = WMMA_F32(a1, b1, q1);
        p2 = WMMA_F32(a0, b2, p2);  q2 = WMMA_F32(a1, b2, q2);
        p3 = WMMA_F32(a0, b3, p3);  q3 = WMMA_F32(a1, b3, q3);
    }

    // C layout: VGPR v holds rows (v, v+8) split across lane halves.
    const int cm0 = blockIdx.x * 32 + 8 * khalf;
    v8f pp[4] = {p0, p1, p2, p3};
    v8f qq[4] = {q0, q1, q2, q3};
#pragma unroll
    for (int sub = 0; sub < 4; ++sub) {
        const int col = blockIdx.y * 64 + sub * 16 + lo;
        const float bv = bias ? bias[col] : 0.0f;
#pragma unroll
        for (int v = 0; v < 8; ++v) {
            float val0 = pp[sub][v] + bv;
            float val1 = qq[sub][v] + bv;
            if (act) { val0 = fmaxf(val0, 0.0f); val1 = fmaxf(val1, 0.0f); }
            C[(size_t)(cm0 + v)      * ldc + coff + col] = val0;
            C[(size_t)(cm0 + 16 + v) * ldc + coff + col] = val1;
        }
    }
}

// =====================================================================
// Weight packer: BT[k*N+n] = (k<K) ? W[n*K+k] : 0     (BT is Kp x N)
// =====================================================================
__global__ __launch_bounds__(256)
void pack_bt_kernel(const float* __restrict__ W, float* __restrict__ BT,
                    int Nn, int K, int Kp)
{
    int idx = blockIdx.x * 256 + threadIdx.x;
    if (idx >= Kp * Nn) return;
    int k = idx / Nn, n = idx % Nn;
    BT[idx] = (k < K) ? W[(size_t)n * K + k] : 0.0f;
}

// =====================================================================
// Fused attention core for one entity type, one chunk.
// 1 wave per row: scores (q.k /8) -> softmax over S -> o = a.V ->
// LayerNorm(256) -> ReLU -> Ct row.  8 rows per 256-thread block.
// =====================================================================
__global__ __launch_bounds__(256)
void attn_kernel(const float* __restrict__ Qm, const float* __restrict__ Kc,
                 const float* __restrict__ Vc, const float* __restrict__ g,
                 const float* __restrict__ bln, float* __restrict__ Ct,
                 int cbase, int S)
{
    __shared__ float shq[8][257];
    __shared__ float sa[8][40];
    const int lane = threadIdx.x & 31;
    const int w    = threadIdx.x >> 5;
    const int lr   = blockIdx.x * 8 + w;    // chunk-local row
    const int gr   = cbase + lr;            // global row

    const float* qrow = Qm + (size_t)gr * 256;
#pragma unroll
    for (int j = 0; j < 8; ++j) shq[w][lane + 32 * j] = qrow[lane + 32 * j];
    __syncthreads();

    const int P = 4 * S;
    for (int p = lane; p < P; p += 32) {
        const int H = p / S, s = p % S;
        const float* kr = Kc + ((size_t)lr * S + s) * 256 + H * 64;
        const float* qh = &shq[w][H * 64];
        float acc = 0.f;
#pragma unroll
        for (int h = 0; h < 64; ++h) acc += qh[h] * kr[h];
        sa[w][p] = acc * 0.125f;            // 1/sqrt(64)
    }
    __syncthreads();

    if (lane < 4) {                         // softmax per head
        float* row = &sa[w][lane * S];
        float mx = row[0];
        for (int s = 1; s < S; ++s) mx = fmaxf(mx, row[s]);
        float sum = 0.f;
        for (int s = 0; s < S; ++s) { float e = __expf(row[s] - mx); row[s] = e; sum += e; }
        float inv = 1.0f / sum;
        for (int s = 0; s < S; ++s) row[s] *= inv;
    }
    __syncthreads();

    float o[8];
#pragma unroll
    for (int j = 0; j < 8; ++j) {
        const int idx = lane + 32 * j;
        const int H = idx >> 6;
        float acc = 0.f;
        for (int s = 0; s < S; ++s)
            acc += sa[w][H * S + s] * Vc[((size_t)lr * S + s) * 256 + idx];
        o[j] = acc;
    }
    // LayerNorm over 256 via wave reductions
    float ps = 0.f;
#pragma unroll
    for (int j = 0; j < 8; ++j) ps += o[j];
#pragma unroll
    for (int m = 16; m; m >>= 1) ps += __shfl_xor(ps, m, 32);
    const float mean = ps * (1.0f / 256.0f);
    float pv = 0.f;
#pragma unroll
    for (int j = 0; j < 8; ++j) { float d = o[j] - mean; pv += d * d; }
#pragma unroll
    for (int m = 16; m; m >>= 1) pv += __shfl_xor(pv, m, 32);
    const float rstd = rsqrtf(pv * (1.0f / 256.0f) + 1e-5f);

    float* crow = Ct + (size_t)gr * 256;
#pragma unroll
    for (int j = 0; j < 8; ++j) {
        const int idx = lane + 32 * j;
        float val = (o[j] - mean) * rstd * g[idx] + bln[idx];
        crow[idx] = fmaxf(val, 0.0f);
    }
}

// =====================================================================
// One GRU scan step. Each wave: 32 batch rows x 16 hidden cols; computes
// the three gate tiles of Gh = (h*m) @ Whh^T via fp32 WMMAs (K=128) with
// each Whh^T fragment reused across both row halves, then fused
// sigmoid/tanh gate math.  Gi already contains x@Wih^T + bih.
// grid = (N/32, D/16), block = 32.
// =====================================================================
__global__ __launch_bounds__(32)
void gru_step_kernel(const float* __restrict__ hprev, const float* __restrict__ mt,
                     const float* __restrict__ Gi, const float* __restrict__ WhhT,
                     const float* __restrict__ bhh, float* __restrict__ hout)
{
    const int l     = threadIdx.x;
    const int lo    = l & 15;
    const int khalf = l >> 4;
    const int m0    = blockIdx.x * 32 + lo;
    const int m1    = m0 + 16;
    const int dbase = blockIdx.y * 16;

    const float mv0 = mt[m0];
    const float mv1 = mt[m1];
    const float* hr0 = hprev + (size_t)m0 * DD;
    const float* hr1 = hprev + (size_t)m1 * DD;

    v8f r0 = {}, z0 = {}, n0 = {};
    v8f r1 = {}, z1 = {}, n1 = {};
    for (int k0 = 0; k0 < DD; k0 += 4) {
        const int ka = k0 + 2 * khalf;
        v2f a0; a0.x = hr0[ka] * mv0; a0.y = hr0[ka + 1] * mv0;
        v2f a1; a1.x = hr1[ka] * mv1; a1.y = hr1[ka + 1] * mv1;
        const float* B0 = WhhT + (size_t)ka * 384;
        const float* B1 = B0 + 384;
        const int c0 = dbase + lo;
        v2f br; br.x = B0[c0];       br.y = B1[c0];
        v2f bz; bz.x = B0[c0 + 128]; bz.y = B1[c0 + 128];
        v2f bn; bn.x = B0[c0 + 256]; bn.y = B1[c0 + 256];
        r0 = WMMA_F32(a0, br, r0);  r1 = WMMA_F32(a1, br, r1);
        z0 = WMMA_F32(a0, bz, z0);  z1 = WMMA_F32(a1, bz, z1);
        n0 = WMMA_F32(a0, bn, n0);  n1 = WMMA_F32(a1, bn, n1);
    }

    const int cm = blockIdx.x * 32 + 8 * khalf;
    const int dc = dbase + lo;
    const float br_ = bhh[dc], bz_ = bhh[128 + dc], bn_ = bhh[256 + dc];
#pragma unroll
    for (int v = 0; v < 8; ++v) {
#pragma unroll
        for (int half = 0; half < 2; ++half) {
            const int row = cm + 16 * half + v;
            const float ar = half ? r1[v] : r0[v];
            const float az = half ? z1[v] : z0[v];
            const float an = half ? n1[v] : n0[v];
            const float* gi = Gi + (size_t)row * 384;
            const float r = 1.f / (1.f + __expf(-(gi[dc]       + ar + br_)));
            const float z = 1.f / (1.f + __expf(-(gi[128 + dc] + az + bz_)));
            const float n = tanhf(gi[256 + dc] + r * (an + bn_));
            const float hm = hprev[(size_t)row * DD + dc] * mt[row];
            hout[(size_t)row * DD + dc] = (1.f - z) * n + z * hm;
        }
    }
}

// value head: wave-per-row dot(hc, clin_W) + b
__global__ __launch_bounds__(256)
void value_kernel(const float* __restrict__ hc, const float* __restrict__ w,
                  const float* __restrict__ b, float* __restrict__ out)
{
    const int lane = threadIdx.x & 31;
    const int row  = blockIdx.x * 8 + (threadIdx.x >> 5);
    const float* hr = hc + (size_t)row * DD;
    float acc = 0.f;
#pragma unroll
    for (int j = 0; j < 4; ++j) acc += hr[lane + 32 * j] * w[lane + 32 * j];
#pragma unroll
    for (int m = 16; m; m >>= 1) acc += __shfl_xor(acc, m, 32);
    if (lane == 0) out[row] = acc + b[0];
}

__global__ __launch_bounds__(256)
void copy_glo_kernel(const float* __restrict__ in, float* __restrict__ feat)
{
    int idx = blockIdx.x * 256 + threadIdx.x;    // R*28
    if (idx >= RR * 28) return;
    int r = idx / 28, c = idx % 28;
    feat[(size_t)r * 348 + 320 + c] = in[(size_t)r * OBSZ + 550 + c];
}

__global__ __launch_bounds__(256)
void copy_kernel(const float* __restrict__ src, float* __restrict__ dst, int n)
{
    int idx = blockIdx.x * 256 + threadIdx.x;
    if (idx < n) dst[idx] = src[idx];
}

// =====================================================================
// Host side
// =====================================================================
struct TypeParams { const float *Wq, *bq, *Wk, *Wv, *Wo, *g, *b; };
struct AllParams {
    const float *inputs, *h0, *masks;
    TypeParams ball, food, virus, oppo, ptnr;
    const float *em_W, *em_b, *gWih, *gWhh, *gbih, *gbhh;
    const float *aW1, *ab1, *aW2, *ab2, *cW1, *cb1, *cW2, *cb2, *clW, *clb;
};

static void resolve(void* const* d_in, const int* in_sizes, int n_in, AllParams& P)
{
    auto F = [&](int i) { return (const float*)d_in[i]; };
    // insertion order -> d_in[1] is rnn_hxs (N*D); sorted pytree -> d_in[1] is masks (T*N)
    const bool insertion = (n_in >= 2 && in_sizes[1] == NNB * DD);
    if (insertion) {
        P.inputs = F(0); P.h0 = F(1); P.masks = F(2);
        int i = 3;
        TypeParams* tps[5] = {&P.ball, &P.food, &P.virus, &P.oppo, &P.ptnr};
        for (int t = 0; t < 5; ++t) {
            TypeParams& tp = *tps[t];
            tp.Wq = F(i++); tp.bq = F(i++); tp.Wk = F(i++); tp.Wv = F(i++);
            tp.Wo = F(i++); tp.g = F(i++);  tp.b  = F(i++);
        }
        P.em_W = F(i++); P.em_b = F(i++);
        P.gWih = F(i++); P.gWhh = F(i++); P.gbih = F(i++); P.gbhh = F(i++);
        P.aW1 = F(i++); P.ab1 = F(i++); P.aW2 = F(i++); P.ab2 = F(i++);
        P.cW1 = F(i++); P.cb1 = F(i++); P.cW2 = F(i++); P.cb2 = F(i++);
        P.clW = F(i++); P.clb = F(i++);
    } else {
        // fully-sorted pytree leaves: inputs, masks, <params sorted>, rnn_hxs
        P.inputs = F(0); P.masks = F(1);
        int i = 2;
        P.aW1 = F(i++); P.aW2 = F(i++); P.ab1 = F(i++); P.ab2 = F(i++);
        auto typ = [&](TypeParams& tp) {
            tp.Wk = F(i++); tp.Wo = F(i++); tp.Wq = F(i++); tp.Wv = F(i++);
            tp.b  = F(i++); tp.bq = F(i++); tp.g  = F(i++);
        };
        typ(P.ball);
        P.clW = F(i++); P.clb = F(i++);
        P.cW1 = F(i++); P.cW2 = F(i++); P.cb1 = F(i++); P.cb2 = F(i++);
        P.em_W = F(i++); P.em_b = F(i++);
        typ(P.food);
        P.gWhh = F(i++); P.gWih = F(i++); P.gbhh = F(i++); P.gbih = F(i++);
        typ(P.oppo); typ(P.ptnr); typ(P.virus);
        P.h0 = F(i++);
    }
}

extern "C" void kernel_launch(void* const* d_in, const int* in_sizes, int n_in,
                              void* d_out, int out_size, void* d_ws, size_t ws_size,
                              hipStream_t stream)
{
    (void)out_size; (void)ws_size;
    AllParams P; resolve(d_in, in_sizes, n_in, P);
    float* out = (float*)d_out;
    float* ws  = (float*)d_ws;

    size_t o = 0;
    auto take = [&](size_t n) { float* p = ws + o; o += n; return p; };
    float* WQT  = take((size_t)5 * 28 * 256);
    float* WKT  = take((size_t)5 * 16 * 256);
    float* WVT  = take((size_t)5 * 16 * 256);
    float* WOT  = take((size_t)5 * 256 * 64);
    float* EMWT = take((size_t)348 * 128);
    float* WIHT = take((size_t)128 * 384);
    float* WHHT = take((size_t)128 * 384);
    float* AW1T = take(128 * 128);
    float* AW2T = take(128 * 128);
    float* CW1T = take(128 * 128);
    float* CW2T = take(128 * 128);
    float* Qm   = take((size_t)RR * 256);
    float* U1   = take((size_t)2 * CHK * 10 * 256);   // Kc+Vc, later Gi
    float* U2   = take((size_t)RR * 256);             // Ct, later ys+buf1
    float* feat = take((size_t)RR * 348);
    float* Xe   = take((size_t)RR * 128);
    float* buf2 = take((size_t)RR * 128);

    float* Kc = U1;
    float* Vc = U1 + (size_t)CHK * 10 * 256;
    float* Gi = U1;                       // reuses K/V region after attention
    float* Ct = U2;                       // live only during attention
    float* ys = U2;                       // GRU outputs (T*N x D)
    float* buf1 = U2 + (size_t)RR * 128;  // MLP hidden

    auto pack = [&](const float* W, float* BT, int Nn, int K, int Kp) {
        int tot = Kp * Nn;
        pack_bt_kernel<<<(tot + 255) / 256, 256, 0, stream>>>(W, BT, Nn, K, Kp);
    };

    const TypeParams* tps[5] = {&P.ball, &P.oppo, &P.ptnr, &P.food, &P.virus};
    const int tbase[5] = {0, 210, 360, 150, 185};
    const int Sarr[5]  = {10, 10, 10, 5, 5};
    const int kszA[5]  = {15, 15, 15, 7, 5};
    const int coffA[5] = {0, 64, 128, 192, 256};

    for (int i = 0; i < 5; ++i) {
        pack(tps[i]->Wq, WQT + (size_t)i * 28 * 256, 256, 28, 28);
        pack(tps[i]->Wk, WKT + (size_t)i * 16 * 256, 256, kszA[i], 16);
        pack(tps[i]->Wv, WVT + (size_t)i * 16 * 256, 256, kszA[i], 16);
        pack(tps[i]->Wo, WOT + (size_t)i * 256 * 64, 64, 256, 256);
    }
    pack(P.em_W, EMWT, 128, 348, 348);
    pack(P.gWih, WIHT, 384, 128, 128);
    pack(P.gWhh, WHHT, 384, 128, 128);
    pack(P.aW1, AW1T, 128, 128, 128);
    pack(P.aW2, AW2T, 128, 128, 128);
    pack(P.cW1, CW1T, 128, 128, 128);
    pack(P.cW2, CW2T, 128, 128, 128);

    auto gemm = [&](const float* A, int lda, int amode, int S, int rs, int tb, int ks, int cb,
                    const float* B, int Nn, float* C, int ldc, int coff,
                    const float* bias, int act, int M, int K) {
        dim3 grid(M / 32, Nn / 64);
        gemm32x64_f32<<<grid, 32, 0, stream>>>(A, lda, amode, S, rs, tb, ks, cb,
                                               B, Nn, C, ldc, coff, bias, act, K);
    };

    // ---- attention blocks (type-by-type, K/V chunked through L2-resident ws) ----
    for (int i = 0; i < 5; ++i) {
        const TypeParams& tp = *tps[i];
        // Q = x_glo @ Wq^T + bq   (R x 256)
        gemm(P.inputs + 550, OBSZ, 0, 0, 0, 0, 0, 0,
             WQT + (size_t)i * 28 * 256, 256, Qm, 256, 0, tp.bq, 0, RR, 28);
        for (int c = 0; c < RR / CHK; ++c) {
            const int cb = c * CHK;
            gemm(P.inputs, 0, 1, Sarr[i], OBSZ, tbase[i], kszA[i], cb,
                 WKT + (size_t)i * 16 * 256, 256, Kc, 256, 0, nullptr, 0, CHK * Sarr[i], 16);
            gemm(P.inputs, 0, 1, Sarr[i], OBSZ, tbase[i], kszA[i], cb,
                 WVT + (size_t)i * 16 * 256, 256, Vc, 256, 0, nullptr, 0, CHK * Sarr[i], 16);
            attn_kernel<<<CHK / 8, 256, 0, stream>>>(Qm, Kc, Vc, tp.g, tp.b, Ct, cb, Sarr[i]);
        }
        // em_t = relu(LN(c)) @ Wo^T  ->  feat columns
        gemm(Ct, 256, 0, 0, 0, 0, 0, 0,
             WOT + (size_t)i * 256 * 64, 64, feat, 348, coffA[i], nullptr, 0, RR, 256);
    }
    copy_glo_kernel<<<(RR * 28 + 255) / 256, 256, 0, stream>>>(P.inputs, feat);

    // ---- embedding + batched GRU input projection ----
    gemm(feat, 348, 0, 0, 0, 0, 0, 0, EMWT, 128, Xe, 128, 0, P.em_b, 1, RR, 348);
    gemm(Xe, 128, 0, 0, 0, 0, 0, 0, WIHT, 384, Gi, 384, 0, P.gbih, 0, RR, 128);

    // ---- sequential GRU scan (16 fused WMMA steps) ----
    for (int t = 0; t < TT; ++t) {
        const float* hprev = (t == 0) ? P.h0 : (ys + (size_t)(t - 1) * NNB * DD);
        gru_step_kernel<<<dim3(NNB / 32, DD / 16), 32, 0, stream>>>(
            hprev, P.masks + (size_t)t * NNB, Gi + (size_t)t * NNB * 384,
            WHHT, P.gbhh, ys + (size_t)t * NNB * DD);
    }

    // ---- critic -> value ----
    gemm(ys,   128, 0, 0, 0, 0, 0, 0, CW1T, 128, buf1, 128, 0, P.cb1, 1, RR, 128);
    gemm(buf1, 128, 0, 0, 0, 0, 0, 0, CW2T, 128, buf2, 128, 0, P.cb2, 1, RR, 128);
    value_kernel<<<RR / 8, 256, 0, stream>>>(buf2, P.clW, P.clb, out);

    // ---- actor (second layer writes straight into d_out) ----
    gemm(ys,   128, 0, 0, 0, 0, 0, 0, AW1T, 128, buf1, 128, 0, P.ab1, 1, RR, 128);
    gemm(buf1, 128, 0, 0, 0, 0, 0, 0, AW2T, 128, out + RR, 128, 0, P.ab2, 1, RR, 128);

    // ---- hT ----
    copy_kernel<<<(NNB * DD + 255) / 256, 256, 0, stream>>>(
        ys + (size_t)(TT - 1) * NNB * DD, out + RR + (size_t)RR * DD, NNB * DD);
}